// GNN_76553497084653
// MI455X (gfx1250) — compile-verified
//
#include <hip/hip_runtime.h>
#include <hip/hip_bf16.h>

typedef __attribute__((ext_vector_type(2))) float v2f;
typedef __attribute__((ext_vector_type(8))) float v8f;

// ---------------------------------------------------------------------------
// deg[i] = 1.0 (self loop)
__global__ void gcn_set_one(float* __restrict__ deg, int n) {
    int i = blockIdx.x * blockDim.x + threadIdx.x;
    if (i < n) deg[i] = 1.0f;
}

// deg[dst[e]] += 1
__global__ void gcn_degree(const int* __restrict__ dst, float* __restrict__ deg, int e) {
    int i = blockIdx.x * blockDim.x + threadIdx.x;
    if (i < e) atomicAdd(&deg[dst[i]], 1.0f);
}

// deg -> rsqrt(deg)  (becomes deg_inv_sqrt)
__global__ void gcn_rsqrt(float* __restrict__ deg, int n) {
    int i = blockIdx.x * blockDim.x + threadIdx.x;
    if (i < n) deg[i] = rsqrtf(deg[i]);
}

// grid-stride zero fill
__global__ void gcn_zero(float* __restrict__ p, long long n) {
    long long i = (long long)blockIdx.x * blockDim.x + threadIdx.x;
    long long stride = (long long)gridDim.x * blockDim.x;
    for (; i < n; i += stride) p[i] = 0.0f;
}

// ---------------------------------------------------------------------------
// Y[N, FOUT] = X[N, FIN] @ W[FIN, FOUT]   via V_WMMA_F32_16X16X4_F32.
// One wave owns one 16-wide column tile; its B fragments (FIN x 16 slice of W)
// are loaded into registers ONCE, then reused across RPW row tiles.
// fp32 A fragment (16x4): lanes 0-15 -> K = kk+{0,1}; lanes 16-31 -> K = kk+{2,3};
// the per-lane pair is contiguous in X -> single b64 load.
// fp32 C/D (16x16): VGPR r: lanes 0-15 row M=r, lanes 16-31 row M=r+8, col = lane&15.
template <int FIN, int FOUT, int RPW>
__global__ __launch_bounds__(256)
void gcn_gemm_wmma(const float* __restrict__ X, const float* __restrict__ W,
                   float* __restrict__ Y, int n) {
    constexpr int KSTEPS = FIN / 4;      // wmmas per tile
    constexpr int COLT   = FOUT / 16;    // column tiles
    const int wave = blockIdx.x * (256 / 32) + (threadIdx.x >> 5);
    const int lane = threadIdx.x & 31;
    const int rowTiles  = (n + 15) >> 4;
    const int rowGroups = (rowTiles + RPW - 1) / RPW;
    if (wave >= rowGroups * COLT) return;      // wave-uniform: EXEC stays full

    const int ct = wave & (COLT - 1);          // column tile (COLT is pow2)
    const int rg = wave / COLT;                // row-tile group
    const int colBase = ct << 4;
    const int hi = lane >> 4;                  // wave half
    const int m  = lane & 15;                  // A row / B,C column within tile

    // B fragments for this column tile: resident in VGPRs for the whole wave.
    v2f b[KSTEPS];
#pragma unroll
    for (int k = 0; k < KSTEPS; ++k) {
        const int ka = k * 4 + 2 * hi;
        b[k].x = W[ka * FOUT + colBase + m];
        b[k].y = W[(ka + 1) * FOUT + colBase + m];
    }

    for (int i = 0; i < RPW; ++i) {
        const int rt = rg * RPW + i;
        if (rt >= rowTiles) break;             // wave-uniform scalar branch
        const int rowBase = rt << 4;
        int rowA = rowBase + m;
        if (rowA > n - 1) rowA = n - 1;        // clamp: loads stay unconditional
        const float* xrow = X + (size_t)rowA * FIN + 2 * hi;

        // pull the next row tile toward the WGP while this one computes
        __builtin_prefetch(xrow + (size_t)16 * FIN, 0, 1);

        v8f c0 = {}, c1 = {};
#pragma unroll
        for (int k = 0; k < KSTEPS; k += 2) {  // KSTEPS is even (16 or 32)
            const v2f a0 = *(const v2f*)(xrow + (k + 0) * 4);  // b64 load
            const v2f a1 = *(const v2f*)(xrow + (k + 1) * 4);  // b64 load
            c0 = __builtin_amdgcn_wmma_f32_16x16x4_f32(false, a0, false, b[k],
                                                       (short)0, c0, false, false);
            c1 = __builtin_amdgcn_wmma_f32_16x16x4_f32(false, a1, false, b[k + 1],
                                                       (short)0, c1, false, false);
        }
        const v8f c = c0 + c1;

        float* yrow = Y + (size_t)rowBase * FOUT + colBase + m;
#pragma unroll
        for (int r = 0; r < 8; ++r) {
            const int row = rowBase + r + 8 * hi;
            if (row < n) yrow[(size_t)(r + 8 * hi) * FOUT] = c[r];
        }
    }
}

// ---------------------------------------------------------------------------
// agg[dst] += xw[src] * (dis[src]*dis[dst]); one thread = one edge x 4 channels.
__global__ void gcn_scatter(const int* __restrict__ src, const int* __restrict__ dst,
                            const float* __restrict__ dis,
                            const float* __restrict__ xw, float* __restrict__ agg,
                            int e, int fout) {
    const int chunks = fout >> 2;
    long long gid = (long long)blockIdx.x * blockDim.x + threadIdx.x;
    long long total = (long long)e * chunks;
    if (gid >= total) return;
    const int ei = (int)(gid / chunks);
    const int f4 = (int)(gid - (long long)ei * chunks) << 2;
    const int s = src[ei];
    const int d = dst[ei];
    const float norm = dis[s] * dis[d];
    const float4 v = *(const float4*)(xw + (long long)s * fout + f4);
    float* o = agg + (long long)d * fout + f4;
    atomicAdd(o + 0, v.x * norm);
    atomicAdd(o + 1, v.y * norm);
    atomicAdd(o + 2, v.z * norm);
    atomicAdd(o + 3, v.w * norm);
}

// agg = relu(agg + xw * dis[i]^2 + b[f])   (in place, becomes next layer's input)
__global__ void gcn_finalize(float* __restrict__ agg, const float* __restrict__ xw,
                             const float* __restrict__ dis, const float* __restrict__ b,
                             int n, int fout) {
    long long gid = (long long)blockIdx.x * blockDim.x + threadIdx.x;
    long long total = (long long)n * fout;
    if (gid >= total) return;
    const int i = (int)(gid / fout);
    const int f = (int)(gid - (long long)i * fout);
    const float dii = dis[i];
    const float v = agg[gid] + xw[gid] * dii * dii + b[f];
    agg[gid] = fmaxf(v, 0.0f);
}

// ---------------------------------------------------------------------------
// pooled sums + counts per graph
__global__ void gcn_pool(const float* __restrict__ h, const int* __restrict__ batch,
                         float* __restrict__ sums, float* __restrict__ cnt, int n, int f) {
    long long gid = (long long)blockIdx.x * blockDim.x + threadIdx.x;
    long long total = (long long)n * f;
    if (gid >= total) return;
    const int i = (int)(gid / f);
    const int ch = (int)(gid - (long long)i * f);
    const int g = batch[i];
    atomicAdd(&sums[(long long)g * f + ch], h[gid]);
    if (ch == 0) atomicAdd(&cnt[g], 1.0f);
}

// pooled = sums/cnt; h = relu(pooled@Wf1+bf1); out = h@Wf2+bf2 — single block.
__global__ __launch_bounds__(256)
void gcn_mlp(const float* __restrict__ sums, const float* __restrict__ cnt,
             const float* __restrict__ Wf1, const float* __restrict__ bf1,
             const float* __restrict__ Wf2, const float* __restrict__ bf2,
             float* __restrict__ out, int G, int F, int H, int O) {
    __shared__ float pooled[64 * 64];
    __shared__ float hid[64 * 32];
    const int tid = threadIdx.x;

    for (int idx = tid; idx < G * F; idx += blockDim.x) {
        const int g = idx / F;
        pooled[idx] = sums[idx] / fmaxf(cnt[g], 1.0f);
    }
    __syncthreads();

    for (int idx = tid; idx < G * H; idx += blockDim.x) {
        const int g = idx / H, j = idx - (idx / H) * H;
        float acc = bf1[j];
        for (int k = 0; k < F; ++k) acc += pooled[g * F + k] * Wf1[k * H + j];
        hid[idx] = fmaxf(acc, 0.0f);
    }
    __syncthreads();

    for (int idx = tid; idx < G * O; idx += blockDim.x) {
        const int g = idx / O, j = idx - (idx / O) * O;
        float acc = bf2[j];
        for (int k = 0; k < H; ++k) acc += hid[g * H + k] * Wf2[k * O + j];
        out[(long long)g * O + j] = acc;
    }
}

// ---------------------------------------------------------------------------
extern "C" void kernel_launch(void* const* d_in, const int* in_sizes, int n_in,
                              void* d_out, int out_size, void* d_ws, size_t ws_size,
                              hipStream_t stream) {
    const float* x    = (const float*)d_in[0];
    const int*   ei   = (const int*)d_in[1];
    const int*   bat  = (const int*)d_in[2];
    const float* W1   = (const float*)d_in[3];
    const float* b1   = (const float*)d_in[4];
    const float* W2   = (const float*)d_in[5];
    const float* b2   = (const float*)d_in[6];
    const float* W3   = (const float*)d_in[7];
    const float* b3   = (const float*)d_in[8];
    const float* Wf1  = (const float*)d_in[9];
    const float* bf1  = (const float*)d_in[10];
    const float* Wf2  = (const float*)d_in[11];
    const float* bf2  = (const float*)d_in[12];

    const int IN_CH = 64, HID = 32, OUT_CH = 200;
    const int N = in_sizes[0] / IN_CH;
    const int E = in_sizes[1] / 2;
    const int G = out_size / OUT_CH;
    const int* srcI = ei;        // edge_index[0, :]
    const int* dstI = ei + E;    // edge_index[1, :]

    // workspace: bufA [N,128] | bufB [N,128] | dis [N] | sums [G,64] | cnt [G]
    float* bufA = (float*)d_ws;
    float* bufB = bufA + (size_t)N * 128;
    float* dis  = bufB + (size_t)N * 128;
    float* sums = dis + N;
    float* cnt  = sums + (size_t)G * 64;

    const int T = 256;
    constexpr int RPW = 4;
    const int rowTiles  = (N + 15) >> 4;
    const int rowGroups = (rowTiles + RPW - 1) / RPW;

    // degree (with self loop) -> deg_inv_sqrt
    gcn_set_one<<<(N + T - 1) / T, T, 0, stream>>>(dis, N);
    gcn_degree<<<(E + T - 1) / T, T, 0, stream>>>(dstI, dis, E);
    gcn_rsqrt<<<(N + T - 1) / T, T, 0, stream>>>(dis, N);

    const float* hin = x;
    const float* Ws[3] = {W1, W2, W3};
    const float* bs[3] = {b1, b2, b3};
    const int fouts[3] = {128, 128, 64};

    for (int l = 0; l < 3; ++l) {
        const int fout = fouts[l];
        const int waves  = rowGroups * (fout / 16);
        const int blocks = (waves + (T / 32) - 1) / (T / 32);

        if (l == 0)
            gcn_gemm_wmma<64, 128, RPW><<<blocks, T, 0, stream>>>(hin, Ws[l], bufA, N);
        else if (l == 1)
            gcn_gemm_wmma<128, 128, RPW><<<blocks, T, 0, stream>>>(hin, Ws[l], bufA, N);
        else
            gcn_gemm_wmma<128, 64, RPW><<<blocks, T, 0, stream>>>(hin, Ws[l], bufA, N);

        const long long tot = (long long)N * fout;
        gcn_zero<<<2048, T, 0, stream>>>(bufB, tot);

        const long long sthreads = (long long)E * (fout >> 2);
        gcn_scatter<<<(unsigned)((sthreads + T - 1) / T), T, 0, stream>>>(
            srcI, dstI, dis, bufA, bufB, E, fout);

        gcn_finalize<<<(unsigned)((tot + T - 1) / T), T, 0, stream>>>(
            bufB, bufA, dis, bs[l], N, fout);

        hin = bufB;
    }

    // global mean pool + MLP head
    gcn_zero<<<64, T, 0, stream>>>(sums, (long long)G * 64 + G);  // sums & cnt contiguous
    gcn_pool<<<(unsigned)(((long long)N * 64 + T - 1) / T), T, 0, stream>>>(
        bufB, bat, sums, cnt, N, 64);
    gcn_mlp<<<1, T, 0, stream>>>(sums, cnt, Wf1, bf1, Wf2, bf2,
                                 (float*)d_out, G, 64, HID, OUT_CH);
}